// GoalNCA_4337916969665
// MI455X (gfx1250) — compile-verified
//
#include <hip/hip_runtime.h>
#include <hip/hip_bf16.h>
#include <math.h>

// CDNA5 / gfx1250: wave32, WMMA F32 16x16x4 (fp32 matrix core, exact precision).
typedef __attribute__((ext_vector_type(2))) float v2f;
typedef __attribute__((ext_vector_type(8))) float v8f;

#define C_CH   16
#define NIN    56
#define HID    64
#define HW     65536   // 256*256
#define WPIX   256
#define WAVES  4       // waves per block
#define PIX_PER_BLOCK (WAVES * 16)
#define DPAD   65      // padded pixel stride for delta tile (bank-conflict avoidance)

__global__ __launch_bounds__(WAVES * 32)
void goalnca_wmma_kernel(const float* __restrict__ state,
                         const float* __restrict__ goal,
                         const float* __restrict__ rand_mask,
                         const float* __restrict__ w1,
                         const float* __restrict__ b1,
                         const float* __restrict__ w2,
                         const float* __restrict__ b2,
                         float* __restrict__ out) {
    __shared__ float w1s[HID * NIN];      // 64x56, row-major [n*56+k]
    __shared__ float b1s[HID];
    __shared__ float w2s[C_CH * HID];     // 16x64, row-major [n*64+k]
    __shared__ float b2s[C_CH];
    __shared__ float xs[WAVES][16][NIN];  // per-wave feature tile (M=pixel, K=feature)
    __shared__ float hs[WAVES][16][HID];  // per-wave hidden tile
    __shared__ float dls[C_CH][DPAD];     // block-wide delta tile [channel][pixel-in-block]

    const int tid  = threadIdx.x;
    const int wave = tid >> 5;
    const int lane = tid & 31;
    const int m    = lane & 15;       // row within 16-pixel tile / N within C-tile
    const int half = lane >> 4;       // 0: lanes 0-15, 1: lanes 16-31

    // ---- stage weights into LDS (whole block) ----
    for (int i = tid; i < HID * NIN; i += WAVES * 32) w1s[i] = w1[i];
    for (int i = tid; i < C_CH * HID; i += WAVES * 32) w2s[i] = w2[i];
    if (tid < HID)  b1s[tid] = b1[tid];
    if (tid < C_CH) b2s[tid] = b2[tid];
    __syncthreads();

    // ---- pixel coordinates: 64 consecutive pixels per block (within one row) ----
    const int p   = blockIdx.x * PIX_PER_BLOCK + wave * 16 + m;
    const int b   = p >> 16;
    const int rem = p & (HW - 1);
    const int y   = rem >> 8;
    const int x   = rem & (WPIX - 1);
    const int ym = (y > 0) ? y - 1 : 0;
    const int yp = (y < WPIX - 1) ? y + 1 : WPIX - 1;
    const int xm = (x > 0) ? x - 1 : 0;
    const int xp = (x < WPIX - 1) ? x + 1 : WPIX - 1;
    const int rT = ym * WPIX, rM = y * WPIX, rB = yp * WPIX;

    // ---- phase 1: perception features. lane halves split the 16 channels ----
    const int c0 = half * 8;
    #pragma unroll
    for (int cc = 0; cc < 8; ++cc) {
        const int c = c0 + cc;
        const float* sp = state + (((size_t)(b * C_CH + c)) << 16);
        const float tl = sp[rT + xm], tm = sp[rT + x], tr = sp[rT + xp];
        const float ml = sp[rM + xm], mc = sp[rM + x], mr = sp[rM + xp];
        const float bl = sp[rB + xm], bm = sp[rB + x], br = sp[rB + xp];
        // cross-correlation with SX, SY (each /8)
        const float gx = ((tr - tl) + 2.0f * (mr - ml) + (br - bl)) * 0.125f;
        const float gy = ((bl - tl) + 2.0f * (bm - tm) + (br - tr)) * 0.125f;
        xs[wave][m][c]      = mc;
        xs[wave][m][16 + c] = gx;
        xs[wave][m][32 + c] = gy;
    }
    if (half == 0) {
        float g = goal[b * HW + rem];
        g = fminf(fmaxf(g, 0.0f), 1.0f);
        const float pi = 3.14159265358979323846f;
        xs[wave][m][48] = g;
        xs[wave][m][49] = g * g;
        xs[wave][m][50] = 1.0f - g;
        xs[wave][m][51] = __sinf(pi * g);
        xs[wave][m][52] = __cosf(2.0f * pi * g);
        xs[wave][m][53] = sqrtf(g);
        xs[wave][m][54] = 4.0f * g * (1.0f - g);
        xs[wave][m][55] = (g > 0.5f) ? 1.0f : 0.0f;
    }
    __syncthreads();

    // ---- phase 2: GEMM1  h(16x64) = X(16x56) * W1^T(56x64), K-steps of 4 ----
    // A 16x4 layout: lane (half*16+m) holds row m, K = k0+2*half .. +1  (2 VGPRs)
    // B 4x16 layout: lane (half*16+n) holds col n, K = k0+2*half .. +1
    v8f acc0 = {}, acc1 = {}, acc2 = {}, acc3 = {};
    #pragma unroll
    for (int k0 = 0; k0 < NIN; k0 += 4) {
        const int kk = k0 + 2 * half;
        v2f a;
        a.x = xs[wave][m][kk];
        a.y = xs[wave][m][kk + 1];
        v2f bf;
        bf.x = w1s[(0 * 16 + m) * NIN + kk];
        bf.y = w1s[(0 * 16 + m) * NIN + kk + 1];
        acc0 = __builtin_amdgcn_wmma_f32_16x16x4_f32(false, a, false, bf, (short)0, acc0, false, false);
        bf.x = w1s[(1 * 16 + m) * NIN + kk];
        bf.y = w1s[(1 * 16 + m) * NIN + kk + 1];
        acc1 = __builtin_amdgcn_wmma_f32_16x16x4_f32(false, a, false, bf, (short)0, acc1, false, false);
        bf.x = w1s[(2 * 16 + m) * NIN + kk];
        bf.y = w1s[(2 * 16 + m) * NIN + kk + 1];
        acc2 = __builtin_amdgcn_wmma_f32_16x16x4_f32(false, a, false, bf, (short)0, acc2, false, false);
        bf.x = w1s[(3 * 16 + m) * NIN + kk];
        bf.y = w1s[(3 * 16 + m) * NIN + kk + 1];
        acc3 = __builtin_amdgcn_wmma_f32_16x16x4_f32(false, a, false, bf, (short)0, acc3, false, false);
    }

    // D layout: VGPR r, lanes 0-15 -> (M=r, N=lane); lanes 16-31 -> (M=r+8, N=lane-16)
    #pragma unroll
    for (int r = 0; r < 8; ++r) {
        const int mm = r + 8 * half;
        hs[wave][mm][ 0 + m] = fmaxf(acc0[r] + b1s[ 0 + m], 0.0f);
        hs[wave][mm][16 + m] = fmaxf(acc1[r] + b1s[16 + m], 0.0f);
        hs[wave][mm][32 + m] = fmaxf(acc2[r] + b1s[32 + m], 0.0f);
        hs[wave][mm][48 + m] = fmaxf(acc3[r] + b1s[48 + m], 0.0f);
    }
    __syncthreads();

    // ---- phase 3: GEMM2  delta(16x16) = H(16x64) * W2^T(64x16) ----
    v8f d = {};
    #pragma unroll
    for (int k0 = 0; k0 < HID; k0 += 4) {
        const int kk = k0 + 2 * half;
        v2f a;
        a.x = hs[wave][m][kk];
        a.y = hs[wave][m][kk + 1];
        v2f bf;
        bf.x = w2s[m * HID + kk];
        bf.y = w2s[m * HID + kk + 1];
        d = __builtin_amdgcn_wmma_f32_16x16x4_f32(false, a, false, bf, (short)0, d, false, false);
    }

    // ---- transpose delta (+bias) through LDS: D-layout -> [channel][pixel] ----
    {
        const float bias = b2s[m];   // lane's channel = m
        #pragma unroll
        for (int r = 0; r < 8; ++r) {
            const int mm = r + 8 * half;            // pixel row within wave tile
            dls[m][wave * 16 + mm] = d[r] + bias;
        }
    }
    __syncthreads();

    // ---- epilogue: fully coalesced. thread walk over (channel, pixel) ----
    // 16 ch * 64 px = 1024 elems; 128 threads * 8 iters. Each 32-lane run covers
    // 32 consecutive pixels of one channel -> 128B contiguous loads/stores.
    const int p0   = blockIdx.x * PIX_PER_BLOCK;
    const int b0   = p0 >> 16;
    const int rem0 = p0 & (HW - 1);
    #pragma unroll
    for (int i = 0; i < 8; ++i) {
        const int e  = i * (WAVES * 32) + tid;   // 0..1023
        const int c  = e >> 6;                   // channel
        const int px = e & 63;                   // pixel within block
        const int rr = rem0 + px;                // same row, same batch image
        const float fire = (rand_mask[b0 * HW + rr] < 0.5f) ? 1.0f : 0.0f;
        const float s    = xs[px >> 4][px & 15][c];   // center state from feature tile
        const float v    = s + dls[c][px] * fire;
        out[(((size_t)(b0 * C_CH + c)) << 16) + rr] = fminf(fmaxf(v, -1.0f), 1.0f);
    }
}

extern "C" void kernel_launch(void* const* d_in, const int* in_sizes, int n_in,
                              void* d_out, int out_size, void* d_ws, size_t ws_size,
                              hipStream_t stream) {
    const float* state     = (const float*)d_in[0];
    const float* goal      = (const float*)d_in[1];
    const float* rand_mask = (const float*)d_in[2];
    const float* w1        = (const float*)d_in[3];
    const float* b1        = (const float*)d_in[4];
    const float* w2        = (const float*)d_in[5];
    const float* b2        = (const float*)d_in[6];
    float* out = (float*)d_out;

    const int total_pix = 16 * HW;                 // 1,048,576
    const int blocks = total_pix / PIX_PER_BLOCK;  // 16,384
    goalnca_wmma_kernel<<<blocks, WAVES * 32, 0, stream>>>(
        state, goal, rand_mask, w1, b1, w2, b2, out);
}